// GCN_74062416052274
// MI455X (gfx1250) — compile-verified
//
#include <hip/hip_runtime.h>

#define NNODES 100000
#define NEDGES 1600000
#define FDIM   128
#define NGRAPH 512
#define HDIM   64
#define NLAYER 3
#define BN_EPS 1e-5f

typedef float v2f __attribute__((ext_vector_type(2)));
typedef float v8f __attribute__((ext_vector_type(8)));

// ---------------- degree / dinv ----------------
__global__ void k_degree(const int* __restrict__ dst, float* __restrict__ deg) {
    int e = blockIdx.x * blockDim.x + threadIdx.x;
    if (e < NEDGES) unsafeAtomicAdd(&deg[dst[e]], 1.0f);
}

__global__ void k_dinv(float* __restrict__ d) {
    int i = blockIdx.x * blockDim.x + threadIdx.x;
    if (i < NNODES) d[i] = rsqrtf(d[i] + 1.0f);
}

// ---------------- GEMM: HW = H * W via fp32 WMMA ----------------
// grid: NNODES/32 blocks, 256 threads (8 waves). Block covers a 32x128 output
// strip. Wave w owns cols [16w,16w+16) and computes TWO 16x16 tiles (row
// halves) off one shared B fragment -> 2 independent accumulator chains and
// half the global B traffic per WMMA. A strip staged in LDS (pad 132 floats
// per row to avoid the stride-128 16-way bank conflict).
__global__ __launch_bounds__(256) void k_gemm_wmma(const float* __restrict__ H,
                                                   const float* __restrict__ W,
                                                   float* __restrict__ HW) {
    __shared__ float sA[32 * 132];
    const int m0 = blockIdx.x * 32;
    const int t  = threadIdx.x;

    // cooperative load of 32x128 fp32 A strip (1024 float4, 4 per thread)
    const float4* src = (const float4*)(H + (size_t)m0 * FDIM);
#pragma unroll
    for (int i = 0; i < 4; ++i) {
        int idx = t + i * 256;          // float4 index within strip
        int r = idx >> 5;               // 32 float4 per row
        int c = idx & 31;
        float4 v = src[idx];
        *(float4*)&sA[r * 132 + c * 4] = v;
    }
    __syncthreads();

    const int lane = t & 31;
    const int wave = t >> 5;
    const int lm   = lane & 15;         // M (for A) / N (for B,D)
    const int sel  = lane >> 4;         // K-half selector
    const int ncol = wave * 16 + lm;    // output column

    const float* a_base0 = &sA[lm * 132 + sel * 2];
    const float* a_base1 = &sA[(lm + 16) * 132 + sel * 2];
    v8f acc0 = {0.f, 0.f, 0.f, 0.f, 0.f, 0.f, 0.f, 0.f};
    v8f acc1 = {0.f, 0.f, 0.f, 0.f, 0.f, 0.f, 0.f, 0.f};

#pragma unroll 4
    for (int k = 0; k < FDIM; k += 4) {
        v2f a0 = *(const v2f*)(a_base0 + k);        // A[lm   ][k+2sel..+1]
        v2f a1 = *(const v2f*)(a_base1 + k);        // A[lm+16][k+2sel..+1]
        v2f b;
        b.x = W[(k + sel * 2 + 0) * FDIM + ncol];   // B[k+2sel  ][ncol]
        b.y = W[(k + sel * 2 + 1) * FDIM + ncol];   // B[k+2sel+1][ncol]
        acc0 = __builtin_amdgcn_wmma_f32_16x16x4_f32(
            false, a0, false, b, (short)0, acc0, false, false);
        acc1 = __builtin_amdgcn_wmma_f32_16x16x4_f32(
            false, a1, false, b, (short)0, acc1, false, false);
    }

    // D: VGPR v, lane -> row (v + 8*sel), col ncol
    float* out0 = HW + (size_t)(m0 + sel * 8) * FDIM + ncol;
    float* out1 = HW + (size_t)(m0 + 16 + sel * 8) * FDIM + ncol;
#pragma unroll
    for (int v = 0; v < 8; ++v) out0[(size_t)v * FDIM] = acc0[v];
#pragma unroll
    for (int v = 0; v < 8; ++v) out1[(size_t)v * FDIM] = acc1[v];
}

// ---------------- edge scatter: one wave per edge ----------------
__global__ void k_scatter(const float* __restrict__ hw, const int* __restrict__ src,
                          const int* __restrict__ dst, const float* __restrict__ dinv,
                          float* __restrict__ agg) {
    int tid = blockIdx.x * blockDim.x + threadIdx.x;   // NEDGES*32 threads
    int e = tid >> 5;
    int c = (tid & 31) * 4;
    if (e >= NEDGES) return;
    int s = src[e], d = dst[e];
    float norm = dinv[s] * dinv[d];
    float4 v = *(const float4*)(hw + (size_t)s * FDIM + c);
    float* o = agg + (size_t)d * FDIM + c;
    unsafeAtomicAdd(o + 0, v.x * norm);
    unsafeAtomicAdd(o + 1, v.y * norm);
    unsafeAtomicAdd(o + 2, v.z * norm);
    unsafeAtomicAdd(o + 3, v.w * norm);
}

// ---------------- self-loop + bias ----------------
__global__ void k_finalize(float* __restrict__ agg, const float* __restrict__ hw,
                           const float* __restrict__ dinv, const float* __restrict__ bias) {
    int tid = blockIdx.x * blockDim.x + threadIdx.x;   // NNODES*32 threads
    int row = tid >> 5;
    int c = (tid & 31) * 4;
    if (row >= NNODES) return;
    float di = dinv[row];
    float s = di * di;
    float4 a = *(float4*)(agg + (size_t)row * FDIM + c);
    float4 h = *(const float4*)(hw + (size_t)row * FDIM + c);
    a.x += h.x * s + bias[c + 0];
    a.y += h.y * s + bias[c + 1];
    a.z += h.z * s + bias[c + 2];
    a.w += h.w * s + bias[c + 3];
    *(float4*)(agg + (size_t)row * FDIM + c) = a;
}

// ---------------- BN column stats (sum, sumsq) ----------------
__global__ void k_stats(const float* __restrict__ h, float* __restrict__ stats) {
    int t = threadIdx.x;
    int col = t & 127;
    int r = blockIdx.x * 2 + (t >> 7);
    float s = 0.f, sq = 0.f;
    for (; r < NNODES; r += gridDim.x * 2) {
        float v = h[(size_t)r * FDIM + col];
        s += v;
        sq += v * v;
    }
    unsafeAtomicAdd(&stats[col], s);
    unsafeAtomicAdd(&stats[FDIM + col], sq);
}

// fold mean/var/gamma/beta into scale+shift (in place over stats)
__global__ void k_bnparams(float* __restrict__ stats, const float* __restrict__ gamma,
                           const float* __restrict__ beta) {
    int c = threadIdx.x;
    float mu  = stats[c] * (1.0f / NNODES);
    float var = stats[FDIM + c] * (1.0f / NNODES) - mu * mu;
    float sc  = gamma[c] * rsqrtf(var + BN_EPS);
    stats[c] = sc;
    stats[FDIM + c] = beta[c] - mu * sc;
}

__global__ void k_bn_relu(const float* __restrict__ agg, const float* __restrict__ stats,
                          float* __restrict__ out) {
    int tid = blockIdx.x * blockDim.x + threadIdx.x;
    int row = tid >> 5;
    int c = (tid & 31) * 4;
    if (row >= NNODES) return;
    float4 v = *(const float4*)(agg + (size_t)row * FDIM + c);
    float4 o;
    o.x = fmaxf(v.x * stats[c + 0] + stats[FDIM + c + 0], 0.f);
    o.y = fmaxf(v.y * stats[c + 1] + stats[FDIM + c + 1], 0.f);
    o.z = fmaxf(v.z * stats[c + 2] + stats[FDIM + c + 2], 0.f);
    o.w = fmaxf(v.w * stats[c + 3] + stats[FDIM + c + 3], 0.f);
    *(float4*)(out + (size_t)row * FDIM + c) = o;
}

// ---------------- pooling ----------------
__global__ void k_count(const int* __restrict__ batch, unsigned* __restrict__ counts) {
    int i = blockIdx.x * blockDim.x + threadIdx.x;
    if (i < NNODES) atomicAdd(&counts[batch[i]], 1u);
}

__global__ void k_pool(const float* __restrict__ h, const int* __restrict__ batch,
                       float* __restrict__ gsum, unsigned* __restrict__ gmax) {
    int tid = blockIdx.x * blockDim.x + threadIdx.x;
    int row = tid >> 5;
    int c = (tid & 31) * 4;
    if (row >= NNODES) return;
    int g = batch[row];
    float4 v = *(const float4*)(h + (size_t)row * FDIM + c);
    float* gs = gsum + (size_t)g * FDIM + c;
    unsigned* gm = gmax + (size_t)g * FDIM + c;
    unsafeAtomicAdd(gs + 0, v.x);
    unsafeAtomicAdd(gs + 1, v.y);
    unsafeAtomicAdd(gs + 2, v.z);
    unsafeAtomicAdd(gs + 3, v.w);
    // post-ReLU values are >= 0, so uint bit-pattern order == value order
    atomicMax(gm + 0, __float_as_uint(v.x));
    atomicMax(gm + 1, __float_as_uint(v.y));
    atomicMax(gm + 2, __float_as_uint(v.z));
    atomicMax(gm + 3, __float_as_uint(v.w));
}

// ---------------- MLP head ----------------
__global__ void k_mlp1(const float* __restrict__ gsum, const unsigned* __restrict__ gmax,
                       const unsigned* __restrict__ counts, const float* __restrict__ w1,
                       const float* __restrict__ b1, float* __restrict__ hid) {
    int t = blockIdx.x * blockDim.x + threadIdx.x;   // NGRAPH*HDIM
    if (t >= NGRAPH * HDIM) return;
    int g = t >> 6, j = t & 63;
    unsigned cu = counts[g];
    float inv = 1.0f / (float)(cu > 0u ? cu : 1u);
    float acc = b1[j];
    const float* zs = gsum + (size_t)g * FDIM;
    const unsigned* zm = gmax + (size_t)g * FDIM;
#pragma unroll 4
    for (int k = 0; k < FDIM; ++k)
        acc += zs[k] * inv * w1[k * HDIM + j];
#pragma unroll 4
    for (int k = 0; k < FDIM; ++k)
        acc += __uint_as_float(zm[k]) * w1[(FDIM + k) * HDIM + j];
    hid[t] = fmaxf(acc, 0.f);
}

__global__ void k_mlp2(const float* __restrict__ hid, const float* __restrict__ w2,
                       const float* __restrict__ b2, float* __restrict__ out) {
    int g = blockIdx.x * blockDim.x + threadIdx.x;
    if (g >= NGRAPH) return;
    float acc = b2[0];
#pragma unroll 8
    for (int j = 0; j < HDIM; ++j) acc += hid[g * HDIM + j] * w2[j];
    out[g] = acc;
}

extern "C" void kernel_launch(void* const* d_in, const int* in_sizes, int n_in,
                              void* d_out, int out_size, void* d_ws, size_t ws_size,
                              hipStream_t stream) {
    const float* x      = (const float*)d_in[0];
    const int*   ei     = (const int*)d_in[1];
    const int*   batch  = (const int*)d_in[2];
    const float* Ws     = (const float*)d_in[3];
    const float* bs     = (const float*)d_in[4];
    const float* gammas = (const float*)d_in[5];
    const float* betas  = (const float*)d_in[6];
    const float* w1     = (const float*)d_in[7];
    const float* b1     = (const float*)d_in[8];
    const float* w2     = (const float*)d_in[9];
    const float* b2     = (const float*)d_in[10];
    float* out = (float*)d_out;

    const int* srcI = ei;
    const int* dstI = ei + NEDGES;

    char* ws = (char*)d_ws;
    size_t off = 0;
    auto alloc = [&](size_t bytes) -> char* {
        char* p = ws + off;
        off += (bytes + 255) & ~(size_t)255;
        return p;
    };
    float*    X      = (float*)alloc((size_t)NNODES * FDIM * 4);
    float*    Y      = (float*)alloc((size_t)NNODES * FDIM * 4);
    float*    dinv   = (float*)alloc((size_t)NNODES * 4);
    float*    stats  = (float*)alloc(2 * FDIM * 4);
    float*    gsum   = (float*)alloc((size_t)NGRAPH * FDIM * 4);
    unsigned* gmax   = (unsigned*)alloc((size_t)NGRAPH * FDIM * 4);
    unsigned* counts = (unsigned*)alloc((size_t)NGRAPH * 4);
    float*    hid    = (float*)alloc((size_t)NGRAPH * HDIM * 4);
    (void)ws_size; (void)in_sizes; (void)n_in; (void)out_size;

    // degrees -> dinv
    hipMemsetAsync(dinv, 0, (size_t)NNODES * 4, stream);
    k_degree<<<(NEDGES + 255) / 256, 256, 0, stream>>>(dstI, dinv);
    k_dinv<<<(NNODES + 255) / 256, 256, 0, stream>>>(dinv);

    // ping-pong: L0: h=x  hw=X agg=Y out=X
    //            L1: h=X  hw=Y agg=X out=Y
    //            L2: h=Y  hw=X agg=Y out=X
    const float* hcur = x;
    float* hwbuf  = X;
    float* aggbuf = Y;
    for (int l = 0; l < NLAYER; ++l) {
        k_gemm_wmma<<<NNODES / 32, 256, 0, stream>>>(hcur, Ws + (size_t)l * FDIM * FDIM, hwbuf);
        hipMemsetAsync(aggbuf, 0, (size_t)NNODES * FDIM * 4, stream);
        k_scatter<<<(NEDGES * 32) / 256, 256, 0, stream>>>(hwbuf, srcI, dstI, dinv, aggbuf);
        k_finalize<<<(NNODES * 32) / 256, 256, 0, stream>>>(aggbuf, hwbuf, dinv, bs + (size_t)l * FDIM);
        hipMemsetAsync(stats, 0, 2 * FDIM * 4, stream);
        k_stats<<<512, 256, 0, stream>>>(aggbuf, stats);
        k_bnparams<<<1, FDIM, 0, stream>>>(stats, gammas + (size_t)l * FDIM, betas + (size_t)l * FDIM);
        k_bn_relu<<<(NNODES * 32) / 256, 256, 0, stream>>>(aggbuf, stats, hwbuf);
        // rotate buffers
        hcur = hwbuf;
        float* tmp = hwbuf; hwbuf = aggbuf; aggbuf = tmp;
    }

    // pooling (hcur points at final node features)
    hipMemsetAsync(gsum, 0, (size_t)NGRAPH * FDIM * 4, stream);
    hipMemsetAsync(gmax, 0, (size_t)NGRAPH * FDIM * 4, stream);
    hipMemsetAsync(counts, 0, (size_t)NGRAPH * 4, stream);
    k_count<<<(NNODES + 255) / 256, 256, 0, stream>>>(batch, counts);
    k_pool<<<(NNODES * 32) / 256, 256, 0, stream>>>(hcur, batch, gsum, gmax);

    // MLP head
    k_mlp1<<<(NGRAPH * HDIM) / 256, 256, 0, stream>>>(gsum, gmax, counts, w1, b1, hid);
    k_mlp2<<<(NGRAPH + 255) / 256, 256, 0, stream>>>(hid, w2, b2, out);
}